// GraphNetwork_79456894976364
// MI455X (gfx1250) — compile-verified
//
#include <hip/hip_runtime.h>

typedef float v2f __attribute__((ext_vector_type(2)));
typedef float v8f __attribute__((ext_vector_type(8)));

#define N_NODES 8192
#define F_IN    512
#define H_DIM   256
#define R_REL   16
#define B_CONV  32
#define L_SEQ   256
#define C_CLS   7
#define E_EDGE  262144
#define NB_BASES 30
#define D_DIM   768   // F + H

// --------------------------- CDNA5 async-copy helpers ----------------------
// LDS byte offset of a shared-memory pointer: the flat aperture maps LDS as
// addr[31:0], so truncation of the generic pointer is the DS address.
__device__ __forceinline__ unsigned lds_off(const void* p)
{
    return (unsigned)(uintptr_t)p;
}
// GLOBAL_LOAD_ASYNC_TO_LDS_B128 (GV mode): per-lane 16B global -> LDS, ASYNCcnt.
__device__ __forceinline__ void async_copy_b128(unsigned lds, const void* g)
{
    asm volatile("global_load_async_to_lds_b128 %0, %1, off"
                 :: "v"(lds), "v"((unsigned long long)(uintptr_t)g)
                 : "memory");
}
// GLOBAL_LOAD_ASYNC_TO_LDS_B32: per-lane 4B (used for the transpose-on-store tile).
__device__ __forceinline__ void async_copy_b32(unsigned lds, const void* g)
{
    asm volatile("global_load_async_to_lds_b32 %0, %1, off"
                 :: "v"(lds), "v"((unsigned long long)(uintptr_t)g)
                 : "memory");
}
__device__ __forceinline__ void wait_async0()
{
    asm volatile("s_wait_asynccnt 0x0" ::: "memory");
}

// ---------------------------------------------------------------------------
// fp32 WMMA GEMM: C[M,Ncol] = A[M,K] @ B (+bias) (+=C) (relu), batch via z.
// B is [K,Ncol] (TRANSB=0) or [Ncol,K] (TRANSB=1).
// 64x64 block tile, 256 threads (8 waves), wave tile 16x32 via two
// V_WMMA_F32_16X16X4_F32 accumulators. Double-buffered LDS filled by
// GLOBAL_LOAD_ASYNC_TO_LDS (tile t+1 DMA overlaps tile t WMMAs).
// REQUIRES: M % 64 == 0 (or epilogue-guarded), Ncol % 64 == 0, K % 32 == 0
//           (true for every call site below), 16B-aligned A/B bases.
// ---------------------------------------------------------------------------
template<bool TRANSB, bool ADDC, bool BIAS, bool RELU>
__global__ void gemm_f32_wmma(const float* __restrict__ A, const float* __restrict__ B,
                              const float* __restrict__ bias, float* __restrict__ C,
                              int M, int Ncol, int K, int lda, int ldb, int ldc,
                              long sA, long sB, long sC)
{
    // stride 36 floats (144B): rows 16B-aligned for B128; banks 36*l+k conflict-free
    // stride 72 floats (288B): rows k,k+2 land 16 banks apart -> halves disjoint
    __shared__ __align__(16) float As[2][64][36];
    __shared__ __align__(16) float Bs[2][32][72];

    const int bm = blockIdx.y * 64;
    const int bn = blockIdx.x * 64;
    A += (long)blockIdx.z * sA;
    B += (long)blockIdx.z * sB;
    C += (long)blockIdx.z * sC;

    const int tid  = threadIdx.x;
    const int lane = tid & 31;
    const int wid  = tid >> 5;
    const int wr   = wid >> 1;        // wave row tile (0..3)
    const int wc   = wid & 1;         // wave col tile (0..1, 32 cols each)
    const int l16  = lane & 15;
    const int half = (lane >> 4) & 1; // ISA: lanes 16-31 carry K+2/K+3 (A,B) / M+8 (D)

    auto issue_tile = [&](int buf, int k0) {
        #pragma unroll
        for (int i = 0; i < 2; ++i) {              // A tile 64x32: 512 x 16B chunks
            int chunk = tid + i * 256;
            int row = chunk >> 3, kq = (chunk & 7) << 2;
            async_copy_b128(lds_off(&As[buf][row][kq]),
                            A + (long)(bm + row) * lda + (k0 + kq));
        }
        if (!TRANSB) {
            #pragma unroll
            for (int i = 0; i < 2; ++i) {          // B tile 32x64: contiguous rows
                int chunk = tid + i * 256;
                int r = chunk >> 4, cq = (chunk & 15) << 2;
                async_copy_b128(lds_off(&Bs[buf][r][cq]),
                                B + (long)(k0 + r) * ldb + (bn + cq));
            }
        } else {
            #pragma unroll
            for (int i = 0; i < 8; ++i) {          // B^T: transpose-on-store, 4B grain
                int el = tid + i * 256;
                int col = el >> 5, k = el & 31;
                async_copy_b32(lds_off(&Bs[buf][k][col]),
                               B + (long)(bn + col) * ldb + (k0 + k));
            }
        }
    };

    v8f acc0 = {0.f,0.f,0.f,0.f,0.f,0.f,0.f,0.f};
    v8f acc1 = acc0;

    issue_tile(0, 0);
    const int T = K >> 5;
    for (int t = 0; t < T; ++t) {
        wait_async0();          // my slices of tile t are in LDS
        __syncthreads();        // everyone's slices in; everyone done reading buf (t+1)&1 from t-1
        if (t + 1 < T) issue_tile((t + 1) & 1, (t + 1) << 5);   // DMA overlaps WMMAs
        const int cur = t & 1;
        #pragma unroll
        for (int kk = 0; kk < 32; kk += 4) {
            v2f a, b0, b1;
            // A frag (16x4 f32): v0 = K+2*half, v1 = K+1+2*half, M = l16
            a.x  = As[cur][wr*16 + l16][kk + 2*half];
            a.y  = As[cur][wr*16 + l16][kk + 2*half + 1];
            // B frag (4x16 f32): v0 = row K+2*half, v1 = row K+1+2*half, N = l16
            b0.x = Bs[cur][kk + 2*half    ][wc*32 + l16];
            b0.y = Bs[cur][kk + 2*half + 1][wc*32 + l16];
            b1.x = Bs[cur][kk + 2*half    ][wc*32 + 16 + l16];
            b1.y = Bs[cur][kk + 2*half + 1][wc*32 + 16 + l16];
            acc0 = __builtin_amdgcn_wmma_f32_16x16x4_f32(false, a, false, b0, (short)0, acc0, false, false);
            acc1 = __builtin_amdgcn_wmma_f32_16x16x4_f32(false, a, false, b1, (short)0, acc1, false, false);
        }
    }

    // D layout: VGPR v -> row M = v + 8*half, col N = l16
    #pragma unroll
    for (int v = 0; v < 8; ++v) {
        int row = bm + wr*16 + 8*half + v;
        if (row >= M) continue;
        int col0 = bn + wc*32 + l16;
        int col1 = col0 + 16;
        {
            float r0 = acc0[v];
            if (BIAS) r0 += bias[col0];
            if (ADDC) r0 += C[(long)row*ldc + col0];
            if (RELU) r0 = fmaxf(r0, 0.f);
            C[(long)row*ldc + col0] = r0;
        }
        {
            float r1 = acc1[v];
            if (BIAS) r1 += bias[col1];
            if (ADDC) r1 += C[(long)row*ldc + col1];
            if (RELU) r1 = fmaxf(r1, 0.f);
            C[(long)row*ldc + col1] = r1;
        }
    }
}

// ---------------------------------------------------------------------------
__global__ void zero_f32(float* p, long n)
{
    long i = (long)blockIdx.x * blockDim.x + threadIdx.x;
    if (i < n) p[i] = 0.f;
}

// Wcat[f, r*H+h] = sum_b comp[r,b] * basis[b,f,h]
__global__ void wcat_kernel(const float* __restrict__ comp, const float* __restrict__ basis,
                            float* __restrict__ Wcat)
{
    int id = blockIdx.x * blockDim.x + threadIdx.x;   // F * R * H = 2,097,152
    if (id >= F_IN * R_REL * H_DIM) return;
    int f  = id >> 12;       // / (R*H)
    int rc = id & 4095;
    int r  = rc >> 8;
    int h  = rc & 255;
    float acc = 0.f;
    #pragma unroll
    for (int b = 0; b < NB_BASES; ++b)
        acc = fmaf(comp[r*NB_BASES + b], basis[((long)b*F_IN + f)*H_DIM + h], acc);
    Wcat[(long)f*(R_REL*H_DIM) + rc] = acc;
}

__global__ void edge_count(const int* __restrict__ ei, const int* __restrict__ et,
                           float* __restrict__ cnt)
{
    int e = blockIdx.x * blockDim.x + threadIdx.x;
    if (e >= E_EDGE) return;
    int dst = ei[E_EDGE + e];
    atomicAdd(&cnt[dst*R_REL + et[e]], 1.f);
}

// out1[dst] += xw[src, etype] / max(cnt[dst,etype],1)   (64 threads/edge, float4 each)
__global__ void scatter_msg(const int* __restrict__ ei, const int* __restrict__ et,
                            const float* __restrict__ cnt, const float* __restrict__ xw,
                            float* __restrict__ out1)
{
    int id = blockIdx.x * blockDim.x + threadIdx.x;   // E * (H/4)
    int e  = id >> 6;
    int hq = (id & 63) << 2;
    if (e >= E_EDGE) return;
    int src = ei[e], dst = ei[E_EDGE + e], r = et[e];
    float inv = 1.f / fmaxf(cnt[dst*R_REL + r], 1.f);
    const float4 m = *(const float4*)&xw[((long)src*R_REL + r)*H_DIM + hq];
    float* o = &out1[(long)dst*H_DIM + hq];
    atomicAdd(o+0, m.x*inv); atomicAdd(o+1, m.y*inv);
    atomicAdd(o+2, m.z*inv); atomicAdd(o+3, m.w*inv);
}

// agg2[dst] += out1[src]
__global__ void scatter_sum(const int* __restrict__ ei, const float* __restrict__ out1,
                            float* __restrict__ agg2)
{
    int id = blockIdx.x * blockDim.x + threadIdx.x;
    int e  = id >> 6;
    int hq = (id & 63) << 2;
    if (e >= E_EDGE) return;
    int src = ei[e], dst = ei[E_EDGE + e];
    const float4 m = *(const float4*)&out1[(long)src*H_DIM + hq];
    float* o = &agg2[(long)dst*H_DIM + hq];
    atomicAdd(o+0, m.x); atomicAdd(o+1, m.y);
    atomicAdd(o+2, m.z); atomicAdd(o+3, m.w);
}

// emotions = concat(x, out2); masked copy (M[t,b] = emotions[b*L+t], mask == umask flat)
__global__ void build_emotions(const float* __restrict__ x, const float* __restrict__ out2,
                               const float* __restrict__ um,
                               float* __restrict__ emo, float* __restrict__ emom)
{
    long id = (long)blockIdx.x * blockDim.x + threadIdx.x;   // N * D
    if (id >= (long)N_NODES * D_DIM) return;
    int n = (int)(id / D_DIM), d = (int)(id % D_DIM);
    float v = (d < F_IN) ? x[(long)n*F_IN + d] : out2[(long)n*H_DIM + (d - F_IN)];
    emo[id]  = v;
    emom[id] = v * um[n];
}

// a = renorm( softmax_s( tanh(S*m_s) ) * m_s ), in place.  block = 256 threads over s.
__global__ void attn_softmax(float* __restrict__ sc, const float* __restrict__ um)
{
    __shared__ float red[256];
    int t = blockIdx.x, b = blockIdx.y, s = threadIdx.x;
    long row = ((long)b*L_SEQ + t) * L_SEQ;
    float ms = um[b*L_SEQ + s];
    float v  = tanhf(sc[row + s] * ms);
    red[s] = v; __syncthreads();
    for (int o = 128; o > 0; o >>= 1) { if (s < o) red[s] = fmaxf(red[s], red[s+o]); __syncthreads(); }
    float vmax = red[0]; __syncthreads();
    float e = expf(v - vmax);
    red[s] = e; __syncthreads();
    for (int o = 128; o > 0; o >>= 1) { if (s < o) red[s] += red[s+o]; __syncthreads(); }
    float p = (e / red[0]) * ms; __syncthreads();
    red[s] = p; __syncthreads();
    for (int o = 128; o > 0; o >>= 1) { if (s < o) red[s] += red[s+o]; __syncthreads(); }
    sc[row + s] = p / red[0];
}

// logits[(t*B+b)*C + c] = hidden[b*L+t] . w_fc[:,c] + b_fc[c]   (one wave32 per node)
__global__ void logits_kernel(const float* __restrict__ hidden, const float* __restrict__ w_fc,
                              const float* __restrict__ b_fc, float* __restrict__ logits)
{
    int wid = threadIdx.x >> 5, lane = threadIdx.x & 31;
    int n = blockIdx.x * 8 + wid;
    if (n >= N_NODES) return;
    float acc[C_CLS];
    #pragma unroll
    for (int c = 0; c < C_CLS; ++c) acc[c] = 0.f;
    for (int k = lane; k < H_DIM; k += 32) {
        float h = hidden[(long)n*H_DIM + k];
        #pragma unroll
        for (int c = 0; c < C_CLS; ++c) acc[c] = fmaf(h, w_fc[k*C_CLS + c], acc[c]);
    }
    #pragma unroll
    for (int c = 0; c < C_CLS; ++c)
        for (int o = 16; o > 0; o >>= 1) acc[c] += __shfl_down(acc[c], o, 32);
    if (lane == 0) {
        int b = n / L_SEQ, t = n % L_SEQ;
        long base = ((long)t*B_CONV + b) * C_CLS;
        #pragma unroll
        for (int c = 0; c < C_CLS; ++c) logits[base + c] = acc[c] + b_fc[c];
    }
}

// Faithful reference: log_softmax(axis=1) on [L,B,C] == normalize over the 32
// conversations. lane = b: exact wave32 fit, one wave per (t,c).
__global__ void logsoftmax_axisB(const float* __restrict__ logits, float* __restrict__ out)
{
    int wid = threadIdx.x >> 5, lane = threadIdx.x & 31;
    int p = blockIdx.x * 8 + wid;          // p = t*C + c, total L*C = 1792
    if (p >= L_SEQ * C_CLS) return;
    int t = p / C_CLS, c = p % C_CLS;
    long idx = ((long)t*B_CONV + lane) * C_CLS + c;
    float v = logits[idx];
    float m = v;
    for (int o = 16; o > 0; o >>= 1) m = fmaxf(m, __shfl_xor(m, o, 32));
    float e = expf(v - m), s = e;
    for (int o = 16; o > 0; o >>= 1) s += __shfl_xor(s, o, 32);
    out[idx] = v - m - logf(s);
}

// ---------------------------------------------------------------------------
extern "C" void kernel_launch(void* const* d_in, const int* in_sizes, int n_in,
                              void* d_out, int out_size, void* d_ws, size_t ws_size,
                              hipStream_t stream)
{
    (void)in_sizes; (void)n_in; (void)out_size; (void)ws_size;
    const float* x      = (const float*)d_in[0];
    const int*   ei     = (const int*)  d_in[1];
    /* d_in[2] edge_norm: unused by reference */
    const int*   et     = (const int*)  d_in[3];
    /* d_in[4] seq_lengths == L; d_in[6] nodal_attn == 1; d_in[7] avec == 0 (static) */
    const float* um     = (const float*)d_in[5];
    const float* basis  = (const float*)d_in[8];
    const float* comp   = (const float*)d_in[9];
    const float* root1  = (const float*)d_in[10];
    const float* bias1  = (const float*)d_in[11];
    const float* w_rel  = (const float*)d_in[12];
    const float* b_rel  = (const float*)d_in[13];
    const float* w_root = (const float*)d_in[14];
    const float* w_att  = (const float*)d_in[15];
    const float* b_att  = (const float*)d_in[16];
    const float* w_lin  = (const float*)d_in[17];
    const float* b_lin  = (const float*)d_in[18];
    const float* w_fc   = (const float*)d_in[19];
    const float* b_fc   = (const float*)d_in[20];
    float* out = (float*)d_out;

    // workspace layout (~161.3 MB). xw's 128MB region is recycled for the
    // attention tensors once the edge scatter has consumed it.
    char* ws   = (char*)d_ws;
    float* xw   = (float*)(ws);                          // [N, R, H]  128 MB
    float* emo  = (float*)(ws);                          // [N, D]      24 MB (reuse)
    float* emom = (float*)(ws + ((size_t)24  << 20));    // [N, D]      24 MB
    float* Xb   = (float*)(ws + ((size_t)48  << 20));    // [N, D]      24 MB
    float* attb = (float*)(ws + ((size_t)72  << 20));    // [N, D]      24 MB
    float* Wcat = (float*)(ws + ((size_t)128 << 20));    // [F, R*H]     8 MB
    float* sc   = Wcat;                                  // [B, L, L]    8 MB (reuse)
    float* out1 = (float*)(ws + ((size_t)136 << 20));    // [N, H]       8 MB
    float* agg2 = (float*)(ws + ((size_t)144 << 20));    // [N, H]       8 MB
    float* hid  = agg2;                                  // [N, H]       (reuse)
    float* out2 = (float*)(ws + ((size_t)152 << 20));    // [N, H]       8 MB
    float* cnt  = (float*)(ws + ((size_t)160 << 20));    // [N*R]      0.5 MB
    float* lg   = (float*)(ws + ((size_t)161 << 20));    // [L,B,C]    224 KB

    const dim3 blk(256);
    const long sLD = (long)L_SEQ * D_DIM;   // batch stride in node-major [N,D] tensors
    const long sLL = (long)L_SEQ * L_SEQ;

    // 0) per-(node,relation) edge counts
    zero_f32<<<(N_NODES*R_REL + 255)/256, blk, 0, stream>>>(cnt, (long)N_NODES*R_REL);
    edge_count<<<E_EDGE/256, blk, 0, stream>>>(ei, et, cnt);

    // 1) Wcat = einsum('rb,bfh') laid out [F, R*H] so RGCN becomes one GEMM
    wcat_kernel<<<(F_IN*R_REL*H_DIM)/256, blk, 0, stream>>>(comp, basis, Wcat);

    // 2) xw = x @ Wcat : 8192 x 4096, K=512 (dominant GEMM, 34 GFLOP)
    gemm_f32_wmma<false,false,false,false><<<dim3(4096/64, 8192/64, 1), blk, 0, stream>>>(
        x, Wcat, nullptr, xw, N_NODES, R_REL*H_DIM, F_IN, F_IN, R_REL*H_DIM, R_REL*H_DIM, 0,0,0);

    // 3) out1 = x @ root1 + bias1
    gemm_f32_wmma<false,false,true,false><<<dim3(256/64, 8192/64, 1), blk, 0, stream>>>(
        x, root1, bias1, out1, N_NODES, H_DIM, F_IN, F_IN, H_DIM, H_DIM, 0,0,0);

    // 4) out1 += mean-normalized relational messages (L2-resident gathers + f32 atomics)
    scatter_msg<<<(E_EDGE*64)/256, blk, 0, stream>>>(ei, et, cnt, xw, out1);

    // 5) agg2 = segment_sum(out1[src], dst)
    zero_f32<<<(N_NODES*H_DIM + 255)/256, blk, 0, stream>>>(agg2, (long)N_NODES*H_DIM);
    scatter_sum<<<(E_EDGE*64)/256, blk, 0, stream>>>(ei, out1, agg2);

    // 6) out2 = agg2 @ w_rel + b_rel ; out2 += out1 @ w_root
    gemm_f32_wmma<false,false,true,false><<<dim3(4, 128, 1), blk, 0, stream>>>(
        agg2, w_rel, b_rel, out2, N_NODES, H_DIM, H_DIM, H_DIM, H_DIM, H_DIM, 0,0,0);
    gemm_f32_wmma<false,true,false,false><<<dim3(4, 128, 1), blk, 0, stream>>>(
        out1, w_root, nullptr, out2, N_NODES, H_DIM, H_DIM, H_DIM, H_DIM, H_DIM, 0,0,0);

    // 7) emotions / masked memory (M[t,b] = emotions[b*L+t]; seq_lengths are all L)
    build_emotions<<<((long)N_NODES*D_DIM)/256, blk, 0, stream>>>(x, out2, um, emo, emom);

    // 8) X = emotions @ w_att + b_att : 8192 x 768, K=768
    gemm_f32_wmma<false,false,true,false><<<dim3(12, 128, 1), blk, 0, stream>>>(
        emo, w_att, b_att, Xb, N_NODES, D_DIM, D_DIM, D_DIM, D_DIM, D_DIM, 0,0,0);

    // 9) scores_b = X_b @ Mm_b^T  (batched NT, 32 x [256x256x768])
    gemm_f32_wmma<true,false,false,false><<<dim3(4, 4, B_CONV), blk, 0, stream>>>(
        Xb, emom, nullptr, sc, L_SEQ, L_SEQ, D_DIM, D_DIM, D_DIM, L_SEQ, sLD, sLD, sLL);

    // 10) a = renorm(softmax(tanh(scores*mask))*mask)
    attn_softmax<<<dim3(L_SEQ, B_CONV), blk, 0, stream>>>(sc, um);

    // 11) att_b = a_b @ M_b  (batched NN), stored node-major [b*L+t, D]
    gemm_f32_wmma<false,false,false,false><<<dim3(12, 4, B_CONV), blk, 0, stream>>>(
        sc, emo, nullptr, attb, L_SEQ, D_DIM, L_SEQ, L_SEQ, D_DIM, D_DIM, sLL, sLD, sLD);

    // 12) hidden = relu(att @ w_lin + b_lin) : 8192 x 256, K=768
    gemm_f32_wmma<false,false,true,true><<<dim3(4, 128, 1), blk, 0, stream>>>(
        attb, w_lin, b_lin, hid, N_NODES, H_DIM, D_DIM, D_DIM, H_DIM, H_DIM, 0,0,0);

    // 13) logits (t-major [L,B,C]) then log_softmax over the batch axis (faithful axis=1)
    logits_kernel<<<N_NODES/8, blk, 0, stream>>>(hid, w_fc, b_fc, lg);
    logsoftmax_axisB<<<(L_SEQ*C_CLS + 7)/8, blk, 0, stream>>>(lg, out);
}